// GATPlusTTrain_35021163331764
// MI455X (gfx1250) — compile-verified
//
#include <hip/hip_runtime.h>
#include <hip/hip_bf16.h>

// ---- problem constants (match reference) ----
#define N0      100000
#define N1      50000
#define NE      250000
#define TT      4
#define D_NODE  128
#define D_EDGE  16
#define D_TIME  32
#define D_MSG   176          // D_NODE + D_EDGE + D_TIME
#define D_CAT   304          // D_MSG + D_NODE
#define N_CLASS 16
#define NEG_SLOPE 0.01f

typedef float v2f __attribute__((ext_vector_type(2)));
typedef float v8f __attribute__((ext_vector_type(8)));

__device__ __forceinline__ float wave_sum32(float p) {
    p += __shfl_xor(p, 16, 32);
    p += __shfl_xor(p, 8, 32);
    p += __shfl_xor(p, 4, 32);
    p += __shfl_xor(p, 2, 32);
    p += __shfl_xor(p, 1, 32);
    return p;
}

__device__ __forceinline__ float leaky(float x) {
    return x >= 0.f ? x : NEG_SLOPE * x;
}

// ---------------------------------------------------------------------------
// K1: one wave per destination node.
// X[n, 0:176]   = -leaky(self_h_tmp . Wa + ba) * self_h_tmp   (self-loop term)
// X[n, 176:304] = self_h
// ---------------------------------------------------------------------------
__global__ void k1_self_init(const float* __restrict__ nf,
                             const float* __restrict__ Wa,
                             const float* __restrict__ phase,
                             const float* __restrict__ ba_p,
                             const int*   __restrict__ layer_nid,
                             float* __restrict__ X) {
    const int n    = (blockIdx.x * blockDim.x + threadIdx.x) >> 5;   // exact: 50000
    const int lane = threadIdx.x & 31;
    const int nid  = layer_nid[n];

    const float4 h   = ((const float4*)(nf + (size_t)nid * D_NODE))[lane];
    const float4 wah = ((const float4*)Wa)[lane];
    const float  te0 = __cosf(phase[lane]);                 // time_encode(0)
    const float  wat = Wa[D_NODE + D_EDGE + lane];          // Wa[144+lane]

    float p = h.x * wah.x + h.y * wah.y + h.z * wah.z + h.w * wah.w + te0 * wat;
    p = wave_sum32(p);
    const float a  = leaky(p + ba_p[0]);
    const float na = -a;

    float* row = X + (size_t)n * D_CAT;
    float4 o0; o0.x = na * h.x; o0.y = na * h.y; o0.z = na * h.z; o0.w = na * h.w;
    ((float4*)row)[lane] = o0;                      // cols 0..127
    if (lane < D_EDGE) row[D_NODE + lane] = 0.f;    // cols 128..143 (zero edge feats)
    row[D_NODE + D_EDGE + lane] = na * te0;         // cols 144..175
    ((float4*)(row + D_MSG))[lane] = h;             // cols 176..303 = self_h
}

// ---------------------------------------------------------------------------
// K2: one wave per edge. 176-dim attention dot per timestep via wave reduce,
// weighted message accumulation, atomic scatter into X[dst].
// ---------------------------------------------------------------------------
__global__ void k2_edge_scatter(const float* __restrict__ nf,
                                const float* __restrict__ ef,
                                const float* __restrict__ st,
                                const float* __restrict__ basis,
                                const float* __restrict__ phase,
                                const float* __restrict__ Wa,
                                const float* __restrict__ ba_p,
                                const int*   __restrict__ src,
                                const int*   __restrict__ dst,
                                float* __restrict__ X) {
    const int e    = (blockIdx.x * blockDim.x + threadIdx.x) >> 5;   // exact: 250000
    const int lane = threadIdx.x & 31;
    const int s = src[e];
    const int d = dst[e];

    const float4 h   = ((const float4*)(nf + (size_t)s * D_NODE))[lane];
    const float4 wah = ((const float4*)Wa)[lane];
    const float  wae = (lane < D_EDGE) ? Wa[D_NODE + lane] : 0.f;
    const float  wat = Wa[D_NODE + D_EDGE + lane];
    const float  fl  = basis[lane];
    const float  ph  = phase[lane];
    const float  ba  = ba_p[0];

    const float hdot = h.x * wah.x + h.y * wah.y + h.z * wah.z + h.w * wah.w;

    float asum = 0.f, acc_e = 0.f, acc_t = 0.f;
#pragma unroll
    for (int t = 0; t < TT; ++t) {
        const float ts  = st[(size_t)e * TT + t];
        const float efv = (lane < D_EDGE)
                        ? ef[((size_t)e * TT + t) * D_EDGE + lane] : 0.f;
        const float te  = __cosf(ts * fl + ph);
        float p = hdot + efv * wae + te * wat;
        p = wave_sum32(p);
        const float a = leaky(p + ba);
        asum  += a;
        acc_e += a * efv;
        acc_t += a * te;
    }

    float* row = X + (size_t)d * D_CAT;
    atomicAdd(row + 4 * lane + 0, asum * h.x);
    atomicAdd(row + 4 * lane + 1, asum * h.y);
    atomicAdd(row + 4 * lane + 2, asum * h.z);
    atomicAdd(row + 4 * lane + 3, asum * h.w);
    if (lane < D_EDGE) atomicAdd(row + D_NODE + lane, acc_e);
    atomicAdd(row + D_NODE + D_EDGE + lane, acc_t);
}

// ---------------------------------------------------------------------------
// K3: hidden = relu(X @ Wn^T + bn) via V_WMMA_F32_16X16X4_F32.
// One wave per 16x16 output tile; tile_m = blockIdx.x, tile_n = wave-in-block.
// f32 A 16x4 layout: lanes0-15 -> K={k,k+1}, lanes16-31 -> K={k+2,k+3}.
// ---------------------------------------------------------------------------
__global__ void k3_gemm1(const float* __restrict__ X,
                         const float* __restrict__ Wn,
                         const float* __restrict__ bn,
                         float* __restrict__ hidden) {
    const int tile_m = blockIdx.x;            // 0..3124
    const int tile_n = threadIdx.x >> 5;      // 0..7
    const int lane   = threadIdx.x & 31;
    const int half   = lane >> 4;             // 0 or 1
    const int lr     = lane & 15;

    const float* arow = X  + (size_t)(tile_m * 16 + lr) * D_CAT + 2 * half;
    const float* brow = Wn + (size_t)(tile_n * 16 + lr) * D_CAT + 2 * half;

    v8f c = {};
#pragma unroll 4
    for (int k = 0; k < D_CAT; k += 4) {
        v2f a = *(const v2f*)(arow + k);
        v2f b = *(const v2f*)(brow + k);
        c = __builtin_amdgcn_wmma_f32_16x16x4_f32(
                false, a, false, b, (short)0, c, false, false);
    }

    const float bias = bn[tile_n * 16 + lr];
#pragma unroll
    for (int v = 0; v < 8; ++v) {
        const int m = tile_m * 16 + v + 8 * half;
        float val = c[v] + bias;
        val = val > 0.f ? val : 0.f;
        hidden[(size_t)m * D_NODE + tile_n * 16 + lr] = val;
    }
}

// ---------------------------------------------------------------------------
// K4: logits = hidden @ Wfc^T + bfc (16 classes => single col tile per wave).
// ---------------------------------------------------------------------------
__global__ void k4_gemm2(const float* __restrict__ hidden,
                         const float* __restrict__ Wfc,
                         const float* __restrict__ bfc,
                         float* __restrict__ logits) {
    const int tile_m = blockIdx.x * (blockDim.x >> 5) + (threadIdx.x >> 5);
    if (tile_m >= N1 / 16) return;            // uniform per-wave exit
    const int lane = threadIdx.x & 31;
    const int half = lane >> 4;
    const int lr   = lane & 15;

    const float* arow = hidden + (size_t)(tile_m * 16 + lr) * D_NODE + 2 * half;
    const float* brow = Wfc    + (size_t)lr * D_NODE + 2 * half;

    v8f c = {};
#pragma unroll 8
    for (int k = 0; k < D_NODE; k += 4) {
        v2f a = *(const v2f*)(arow + k);
        v2f b = *(const v2f*)(brow + k);
        c = __builtin_amdgcn_wmma_f32_16x16x4_f32(
                false, a, false, b, (short)0, c, false, false);
    }

    const float bias = bfc[lr];
#pragma unroll
    for (int v = 0; v < 8; ++v) {
        const int m = tile_m * 16 + v + 8 * half;
        logits[(size_t)m * N_CLASS + lr] = c[v] + bias;
    }
}

extern "C" void kernel_launch(void* const* d_in, const int* in_sizes, int n_in,
                              void* d_out, int out_size, void* d_ws, size_t ws_size,
                              hipStream_t stream) {
    const float* nf        = (const float*)d_in[0];   // [N0,128]
    const float* ef        = (const float*)d_in[1];   // [E,T,16]
    const float* st        = (const float*)d_in[2];   // [E,T]
    const float* basis     = (const float*)d_in[3];   // [32]
    const float* phase     = (const float*)d_in[4];   // [32]
    const float* Wa        = (const float*)d_in[5];   // [176]
    const float* ba        = (const float*)d_in[6];   // [1]
    const float* Wn        = (const float*)d_in[7];   // [128,304]
    const float* bn        = (const float*)d_in[8];   // [128]
    const float* Wfc       = (const float*)d_in[9];   // [16,128]
    const float* bfc       = (const float*)d_in[10];  // [16]
    const int*   src       = (const int*)d_in[11];    // [E]
    const int*   dst       = (const int*)d_in[12];    // [E]
    const int*   layer_nid = (const int*)d_in[13];    // [N1]

    float* X      = (float*)d_ws;                      // [N1,304]  60.8 MB
    float* hidden = X + (size_t)N1 * D_CAT;            // [N1,128]  25.6 MB
    float* logits = (float*)d_out;                     // [N1,16]

    // K1: init X with self-loop term + self_h   (50000 waves)
    k1_self_init<<<N1 / 8, 256, 0, stream>>>(nf, Wa, phase, ba, layer_nid, X);

    // K2: edge messages + atomic scatter        (250000 waves)
    k2_edge_scatter<<<NE / 8, 256, 0, stream>>>(nf, ef, st, basis, phase, Wa, ba,
                                                src, dst, X);

    // K3: fused GEMM1 + bias + relu             (3125 x 8 tiles)
    k3_gemm1<<<N1 / 16, 256, 0, stream>>>(X, Wn, bn, hidden);

    // K4: GEMM2 + bias                          (3125 tiles)
    k4_gemm2<<<(N1 / 16 + 7) / 8, 256, 0, stream>>>(hidden, Wfc, bfc, logits);
}